// MPNN_77309411697
// MI455X (gfx1250) — compile-verified
//
#include <hip/hip_runtime.h>
#include <hip/hip_bf16.h>

typedef float v2f __attribute__((ext_vector_type(2)));
typedef float v8f __attribute__((ext_vector_type(8)));

#define B_   2
#define N_   512
#define FN_  128
#define FE_  16
#define MID_ 128

// ---------------- async-to-LDS support (gfx1250) ----------------
#if __has_builtin(__builtin_amdgcn_global_load_async_to_lds_b128) && \
    __has_builtin(__builtin_amdgcn_global_load_async_to_lds_b32)
#define USE_ASYNC_LDS 1
#else
#define USE_ASYNC_LDS 0
#endif

#if USE_ASYNC_LDS
// Parameter types per hipcc diagnostic:
//   b128: (int __vector_size(16) * [global], ... [lds], int offset, int cpol)
//   b32 : (int *                 [global], ... [lds], int offset, int cpol)
typedef int v4i_vs __attribute__((vector_size(16)));
typedef __attribute__((address_space(1))) v4i_vs* gp128_t;
typedef __attribute__((address_space(3))) v4i_vs* lp128_t;
typedef __attribute__((address_space(1))) int*    gp32_t;
typedef __attribute__((address_space(3))) int*    lp32_t;

__device__ __forceinline__ void async_cp_b128(const float* g, float* l) {
    __builtin_amdgcn_global_load_async_to_lds_b128((gp128_t)(void*)(float*)g,
                                                   (lp128_t)(void*)l, 0, 0);
}
__device__ __forceinline__ void async_cp_b32(const float* g, float* l) {
    __builtin_amdgcn_global_load_async_to_lds_b32((gp32_t)(void*)(float*)g,
                                                  (lp32_t)(void*)l, 0, 0);
}
template <int N>
__device__ __forceinline__ void wait_asynccnt() {
#if __has_builtin(__builtin_amdgcn_s_wait_asynccnt)
    __builtin_amdgcn_s_wait_asynccnt(N);
#else
    asm volatile("s_wait_asynccnt %0" ::"i"(N));
#endif
}
#endif

// ---------------------------------------------------------------------------
// msg_g = g_features @ Wg + bg   -> (2,128)   (tiny, one block)
// ---------------------------------------------------------------------------
__global__ void mpnn_gvec(const float* __restrict__ g, const float* __restrict__ Wg,
                          const float* __restrict__ bg, float* __restrict__ out) {
    int t = threadIdx.x;            // 256 threads: (b,m)
    int b = t >> 7, m = t & 127;
    float acc = bg[m];
    const float* gr = g + b * FN_;
    #pragma unroll 8
    for (int k = 0; k < FN_; ++k) acc += gr[k] * Wg[k * MID_ + m];
    out[b * MID_ + m] = acc;
}

// ---------------------------------------------------------------------------
// Generic fp32 WMMA GEMM: out[r,c] = sum_k A[r,k]*W[k,c] + bias[c]
//                                    (+ bcast[batch(r),c]) (+ addmat[r,c])
// A: R x 128, W: 128 x 128. Grid: R/16 blocks, 256 threads (8 waves).
// ---------------------------------------------------------------------------
__global__ void mpnn_gemm(const float* __restrict__ A,
                          const float* __restrict__ W,
                          const float* __restrict__ bias,
                          const float* __restrict__ bcast,
                          const float* __restrict__ addmat,
                          float* __restrict__ out) {
    __shared__ float ldsA[16 * 128];
    const int tid  = threadIdx.x;
    const int lane = tid & 31, wv = tid >> 5;
    const int half = lane >> 4, lc = lane & 15;
    const int rowbase = blockIdx.x * 16;

    #pragma unroll
    for (int s = 0; s < 8; ++s) {
        int q = tid + 256 * s;
        ldsA[q] = A[rowbase * 128 + q];
    }
    __syncthreads();

    const int col = wv * 16 + lc;
    float base = bias[col];
    if (bcast) base += bcast[((rowbase >> 9) << 7) + col];

    v8f c;
    #pragma unroll
    for (int r = 0; r < 8; ++r) {
        float v = base;
        if (addmat) v += addmat[(rowbase + r + 8 * half) * 128 + col];
        c[r] = v;
    }

    #pragma unroll 1
    for (int kc = 0; kc < 8; ++kc) {
        const int k0 = kc * 16;
        #pragma unroll
        for (int t4 = 0; t4 < 4; ++t4) {
            const int kb = k0 + t4 * 4 + 2 * half;   // A: K = vgpr + 2*laneHalf
            v2f a, b;
            a[0] = ldsA[lc * 128 + kb + 0];
            a[1] = ldsA[lc * 128 + kb + 1];
            b[0] = W[(kb + 0) * 128 + col];
            b[1] = W[(kb + 1) * 128 + col];
            c = __builtin_amdgcn_wmma_f32_16x16x4_f32(false, a, false, b,
                                                      (short)0, c, false, false);
        }
    }

    #pragma unroll
    for (int r = 0; r < 8; ++r)
        out[(rowbase + r + 8 * half) * 128 + col] = c[r];
}

// ---------------------------------------------------------------------------
// Fused edge-GEMM + broadcast-sum + masked max over i:
//   msgs[b,j,m] = max_i adj[b,i,j] * (msg1g[b,j,m] + msg2[b,i,m]
//                                      + e[b,i,j,:]@We[:,m] + be[m])
// Grid: (N/16 j-tiles, B). 256 threads = 8 waves; wave w owns m-tile 16w..16w+15.
// Double-buffered LDS staging via GLOBAL_LOAD_ASYNC_TO_LDS (ASYNCcnt pipeline):
// per chunk each thread issues 7 async ops (4xB128 e, 2xB128 msg2, 1xB32 adj),
// so s_wait_asynccnt<=7 after issuing chunk k+1 guarantees chunk k has landed.
// ---------------------------------------------------------------------------
__global__ void mpnn_main(const float* __restrict__ efeat, // (B,N,N,FE)
                          const float* __restrict__ adj,   // (B,N,N)
                          const float* __restrict__ We,    // (FE,MID)
                          const float* __restrict__ be,    // (MID)
                          const float* __restrict__ msg1g, // (B*N,MID) msg1+msg_g
                          const float* __restrict__ msg2,  // (B*N,MID)
                          float* __restrict__ msgs) {      // (B*N,MID)
    __shared__ float ldsE[2][16 * 16 * FE_];   // 2 x 16 KB
    __shared__ float ldsM2[2][16 * MID_];      // 2 x 8 KB
    __shared__ float ldsAdj[2][16 * 16];       // 2 x 1 KB

    const int tid  = threadIdx.x;
    const int lane = tid & 31, wv = tid >> 5;
    const int half = lane >> 4, lc = lane & 15;
    const int b  = blockIdx.y;
    const int j0 = blockIdx.x * 16;
    const int col = wv * 16 + lc;

    const float bias_v = be[col];

    // We column tile held in VGPRs for the whole kernel (4 chains x v2f)
    v2f breg[4];
    #pragma unroll
    for (int t4 = 0; t4 < 4; ++t4) {
        const int kb = t4 * 4 + 2 * half;
        breg[t4][0] = We[(kb + 0) * MID_ + col];
        breg[t4][1] = We[(kb + 1) * MID_ + col];
    }

    // msg1g tile (+be) in C layout, loaded once
    v8f m1;
    #pragma unroll
    for (int r = 0; r < 8; ++r)
        m1[r] = msg1g[(b * N_ + j0 + r + 8 * half) * MID_ + col] + bias_v;

    v8f acc;
    #pragma unroll
    for (int r = 0; r < 8; ++r) acc[r] = -__builtin_inff();

    // ---- staging helpers (per i-chunk of 16) ----
    const float* eBase  = efeat + (size_t)b * N_ * N_ * FE_ + (size_t)j0 * FE_;
    const float* aBase  = adj   + (size_t)b * N_ * N_ + j0;
    const float* m2Base = msg2  + (size_t)b * N_ * MID_;

#if USE_ASYNC_LDS
    auto issue_chunk = [&](int i0, int buf) {
        // e tile: 4096 floats, 4 x b128 per thread; q4 = (tid + 256*s)*4
        #pragma unroll
        for (int s = 0; s < 4; ++s) {
            int q4 = (tid + 256 * s) * 4;
            int ii = q4 >> 8, rem = q4 & 255;
            async_cp_b128(eBase + (size_t)(i0 + ii) * N_ * FE_ + rem,
                          &ldsE[buf][q4]);
        }
        // msg2 chunk: 2048 floats, 2 x b128 per thread
        #pragma unroll
        for (int s = 0; s < 2; ++s) {
            int q4 = (tid + 256 * s) * 4;
            int ii = q4 >> 7, m = q4 & 127;
            async_cp_b128(m2Base + (size_t)(i0 + ii) * MID_ + m,
                          &ldsM2[buf][q4]);
        }
        // adj tile: 256 floats, 1 x b32 per thread
        {
            int ii = tid >> 4, jr = tid & 15;
            async_cp_b32(aBase + (size_t)(i0 + ii) * N_ + jr,
                         &ldsAdj[buf][tid]);
        }
    };
#else
    auto issue_chunk = [&](int i0, int buf) {
        #pragma unroll
        for (int s = 0; s < 16; ++s) {
            int q = tid + 256 * s;
            int ii = q >> 8, rem = q & 255;
            ldsE[buf][q] = eBase[(size_t)(i0 + ii) * N_ * FE_ + rem];
        }
        #pragma unroll
        for (int s = 0; s < 8; ++s) {
            int q = tid + 256 * s;
            int ii = q >> 7, m = q & 127;
            ldsM2[buf][q] = m2Base[(size_t)(i0 + ii) * MID_ + m];
        }
        {
            int ii = tid >> 4, jr = tid & 15;
            ldsAdj[buf][tid] = aBase[(size_t)(i0 + ii) * N_ + jr];
        }
    };
#endif

    issue_chunk(0, 0);

    for (int ic = 0; ic < N_ / 16; ++ic) {
        const int cur = ic & 1;
        const bool more = (ic + 1) < (N_ / 16);
        if (more) issue_chunk((ic + 1) * 16, cur ^ 1);
#if USE_ASYNC_LDS
        if (more) wait_asynccnt<7>(); else wait_asynccnt<0>();
#endif
        __syncthreads();           // chunk `ic` visible to all waves

        #pragma unroll 1
        for (int ii = 0; ii < 16; ++ii) {
            const float m2v = ldsM2[cur][ii * MID_ + col];
            v8f c;
            #pragma unroll
            for (int r = 0; r < 8; ++r) c[r] = m1[r] + m2v;

            const float* ePtr = &ldsE[cur][ii * 256 + lc * FE_];
            #pragma unroll
            for (int t4 = 0; t4 < 4; ++t4) {
                const int kb = t4 * 4 + 2 * half;
                v2f a;
                a[0] = ePtr[kb + 0];
                a[1] = ePtr[kb + 1];
                c = __builtin_amdgcn_wmma_f32_16x16x4_f32(false, a, false, breg[t4],
                                                          (short)0, c, false, false);
            }
            #pragma unroll
            for (int r = 0; r < 8; ++r) {
                const float av = ldsAdj[cur][ii * 16 + r + 8 * half];
                acc[r] = fmaxf(acc[r], av * c[r]);
            }
        }
        __syncthreads();           // WAR: all waves done with buf `cur`
    }

    #pragma unroll
    for (int r = 0; r < 8; ++r)
        msgs[(b * N_ + j0 + r + 8 * half) * MID_ + col] = acc[r];
}

// ---------------------------------------------------------------------------
extern "C" void kernel_launch(void* const* d_in, const int* in_sizes, int n_in,
                              void* d_out, int out_size, void* d_ws, size_t ws_size,
                              hipStream_t stream) {
    const float* features = (const float*)d_in[0];
    const float* efeat    = (const float*)d_in[1];
    const float* gfeat    = (const float*)d_in[2];
    const float* adj      = (const float*)d_in[3];
    const float* W1  = (const float*)d_in[4];
    const float* b1  = (const float*)d_in[5];
    const float* W2  = (const float*)d_in[6];
    const float* b2  = (const float*)d_in[7];
    const float* We  = (const float*)d_in[8];
    const float* be  = (const float*)d_in[9];
    const float* Wg  = (const float*)d_in[10];
    const float* bg  = (const float*)d_in[11];
    const float* Wo1 = (const float*)d_in[12];
    const float* bo1 = (const float*)d_in[13];
    const float* Wo2 = (const float*)d_in[14];
    const float* bo2 = (const float*)d_in[15];

    float* ws = (float*)d_ws;
    const int RM = B_ * N_ * MID_;          // 131072 floats per (B*N,128) buffer
    float* msg1g = ws;                      // msg1 + msg_g broadcast
    float* msg2  = ws + RM;
    float* h1    = ws + 2 * RM;
    float* msgs  = ws + 3 * RM;
    float* gv    = ws + 4 * RM;             // (2,128)

    const int ROWT = (B_ * N_) / 16;        // 64 row-tiles

    mpnn_gvec<<<1, 256, 0, stream>>>(gfeat, Wg, bg, gv);
    mpnn_gemm<<<ROWT, 256, 0, stream>>>(features, W1, b1, gv, nullptr, msg1g);
    mpnn_gemm<<<ROWT, 256, 0, stream>>>(features, W2, b2, nullptr, nullptr, msg2);
    mpnn_gemm<<<ROWT, 256, 0, stream>>>(features, Wo1, bo1, nullptr, nullptr, h1);

    dim3 grid(N_ / 16, B_);
    mpnn_main<<<grid, 256, 0, stream>>>(efeat, adj, We, be, msg1g, msg2, msgs);

    mpnn_gemm<<<ROWT, 256, 0, stream>>>(msgs, Wo2, bo2, nullptr, h1, (float*)d_out);
}